// SimpleGCN_72232759984506
// MI455X (gfx1250) — compile-verified
//
#include <hip/hip_runtime.h>
#include <hip/hip_bf16.h>
#include <math.h>

// ---------------------------------------------------------------------------
// Graph-diffusion GCN for MI455X (gfx1250, wave32).
//   h  = relu(x @ W1) @ W2                       (WMMA f32 16x16x4)
//   4x: g  = scatter_row( w_ns * s1*exp(-|d|^2/2s2^2) * (x[col]-x[row]) )
//       Ax = scatter_row( w_sl * xk[col] ) + wself*xk   (self-loop folded)
//       Gx = scatter_row( w_sl * g[col] )  + wself*g
//       xk = a*h + (1-a-b)*xk + b*Ax + b*g*Gx
//   out = log_softmax(relu(xk) @ W3)
// ---------------------------------------------------------------------------

#define FDIM 64
#define CDIM 40
#define ALPHA 0.1f
#define BETA 0.7f
#define GAMMA 0.2f
#define SIGMA1 0.5f
#define INV2S2 0.5f      // 1/(2*sigma2^2), sigma2=1.0
#define REST 0.2f        // 1 - ALPHA - BETA

typedef __attribute__((ext_vector_type(2))) float v2f;
typedef __attribute__((ext_vector_type(8))) float v8f;

// -------------------------------- utility ----------------------------------

__global__ void zero_kernel(float* __restrict__ p, size_t n) {
  size_t i = (size_t)blockIdx.x * blockDim.x + threadIdx.x;
  size_t stride = (size_t)gridDim.x * blockDim.x;
  for (; i < n; i += stride) p[i] = 0.0f;
}

__global__ void degree_kernel(const int* __restrict__ col, float* __restrict__ deg, int E) {
  int i = blockIdx.x * blockDim.x + threadIdx.x;
  if (i < E) unsafeAtomicAdd(&deg[col[i]], 1.0f);
}

__global__ void dinv_kernel(const float* __restrict__ deg, float* __restrict__ dns,
                            float* __restrict__ dsl, int N) {
  int i = blockIdx.x * blockDim.x + threadIdx.x;
  if (i >= N) return;
  float d = deg[i];
  dns[i] = d > 0.0f ? rsqrtf(d) : 0.0f;
  dsl[i] = rsqrtf(d + 1.0f);          // self-loop adds 1; always > 0
}

__global__ void weights_kernel(const int* __restrict__ row, const int* __restrict__ col,
                               const float* __restrict__ dns, const float* __restrict__ dsl,
                               float* __restrict__ wns, float* __restrict__ wsl, int E) {
  int i = blockIdx.x * blockDim.x + threadIdx.x;
  if (i >= E) return;
  int r = row[i], c = col[i];
  wns[i] = dns[r] * dns[c];
  wsl[i] = dsl[r] * dsl[c];
}

// ------------------------- WMMA MLP: h = relu(xW1)W2 -----------------------
// One wave per 16-row tile. A layout (16x4 f32): lanes 0-15 -> K={0,1},
// lanes 16-31 -> K={2,3}. B mirrored. C/D: vgpr r -> M=r(+8 for hi lanes).

__global__ void __launch_bounds__(128) mlp_kernel(
    const float* __restrict__ x, const float* __restrict__ W1,
    const float* __restrict__ W2, float* __restrict__ h, int N) {
  __shared__ float tile[4][16 * FDIM];
  const int wave = threadIdx.x >> 5;
  const int lane = threadIdx.x & 31;
  const int row0 = (blockIdx.x * 4 + wave) * 16;
  if (row0 >= N) return;

  float* ts = tile[wave];
  // Stage 16x64 f32 tile with coalesced b128 loads.
  const float4* xv = (const float4*)(x + (size_t)row0 * FDIM);
  float4* tv = (float4*)ts;
#pragma unroll
  for (int i = 0; i < 8; ++i) tv[lane + i * 32] = xv[lane + i * 32];
  asm volatile("s_wait_dscnt 0x0" ::: "memory");

  const int m = lane & 15;
  const int hi = lane >> 4;

  // GEMM 1: tmp = x @ W1  (K = 64 in 16 steps of 4), 4 N-tiles of 16.
  v8f acc[4];
#pragma unroll
  for (int t = 0; t < 4; ++t) acc[t] = (v8f)(0.0f);
#pragma unroll
  for (int ks = 0; ks < 16; ++ks) {
    const int k = ks * 4 + hi * 2;
    v2f a;
    a.x = ts[m * FDIM + k];
    a.y = ts[m * FDIM + k + 1];
#pragma unroll
    for (int t = 0; t < 4; ++t) {
      v2f b;
      b.x = W1[k * FDIM + t * 16 + m];
      b.y = W1[(k + 1) * FDIM + t * 16 + m];
      acc[t] = __builtin_amdgcn_wmma_f32_16x16x4_f32(false, a, false, b,
                                                     (short)0, acc[t], false, false);
    }
  }

  // relu + restage tmp into LDS (same-wave LDS ops are in-order; WAR is safe).
#pragma unroll
  for (int t = 0; t < 4; ++t) {
#pragma unroll
    for (int r = 0; r < 8; ++r) {
      float v = acc[t][r];
      v = v > 0.0f ? v : 0.0f;
      ts[(r + hi * 8) * FDIM + t * 16 + m] = v;
    }
  }
  asm volatile("s_wait_dscnt 0x0" ::: "memory");

  // GEMM 2: h = tmp @ W2.
  v8f acc2[4];
#pragma unroll
  for (int t = 0; t < 4; ++t) acc2[t] = (v8f)(0.0f);
#pragma unroll
  for (int ks = 0; ks < 16; ++ks) {
    const int k = ks * 4 + hi * 2;
    v2f a;
    a.x = ts[m * FDIM + k];
    a.y = ts[m * FDIM + k + 1];
#pragma unroll
    for (int t = 0; t < 4; ++t) {
      v2f b;
      b.x = W2[k * FDIM + t * 16 + m];
      b.y = W2[(k + 1) * FDIM + t * 16 + m];
      acc2[t] = __builtin_amdgcn_wmma_f32_16x16x4_f32(false, a, false, b,
                                                      (short)0, acc2[t], false, false);
    }
  }
#pragma unroll
  for (int t = 0; t < 4; ++t) {
#pragma unroll
    for (int r = 0; r < 8; ++r) {
      h[(size_t)(row0 + r + hi * 8) * FDIM + t * 16 + m] = acc2[t][r];
    }
  }
}

// ------------------- per-edge gradient scatter (wave/edge) -----------------

__global__ void grad_kernel(const float* __restrict__ xk, const int* __restrict__ row,
                            const int* __restrict__ col, const float* __restrict__ wns,
                            float* __restrict__ g, int E) {
  const int lane = threadIdx.x & 31;
  int wid = (blockIdx.x * blockDim.x + threadIdx.x) >> 5;
  const int nw = (gridDim.x * blockDim.x) >> 5;
  for (int e = wid; e < E; e += nw) {
    const int r = row[e], c = col[e];
    const float2 xc = *(const float2*)(xk + (size_t)c * FDIM + lane * 2);
    const float2 xr = *(const float2*)(xk + (size_t)r * FDIM + lane * 2);
    const float dx = xc.x - xr.x;
    const float dy = xc.y - xr.y;
    float n2 = dx * dx + dy * dy;
#pragma unroll
    for (int off = 16; off > 0; off >>= 1) n2 += __shfl_xor(n2, off, 32);
    const float coef = wns[e] * (SIGMA1 * __expf(-n2 * INV2S2));
    float* gp = g + (size_t)r * FDIM + lane * 2;
    unsafeAtomicAdd(gp, coef * dx);
    unsafeAtomicAdd(gp + 1, coef * dy);
  }
}

// ---------------- per-edge SpMM scatter: Ax, Gx (wave/edge) ----------------

__global__ void spmm_kernel(const float* __restrict__ xk, const float* __restrict__ g,
                            const int* __restrict__ row, const int* __restrict__ col,
                            const float* __restrict__ wsl, float* __restrict__ Ax,
                            float* __restrict__ Gx, int E) {
  const int lane = threadIdx.x & 31;
  int wid = (blockIdx.x * blockDim.x + threadIdx.x) >> 5;
  const int nw = (gridDim.x * blockDim.x) >> 5;
  for (int e = wid; e < E; e += nw) {
    const int r = row[e], c = col[e];
    const float w = wsl[e];
    const float2 xc = *(const float2*)(xk + (size_t)c * FDIM + lane * 2);
    const float2 gc = *(const float2*)(g + (size_t)c * FDIM + lane * 2);
    const size_t off = (size_t)r * FDIM + lane * 2;
    unsafeAtomicAdd(Ax + off, w * xc.x);
    unsafeAtomicAdd(Ax + off + 1, w * xc.y);
    unsafeAtomicAdd(Gx + off, w * gc.x);
    unsafeAtomicAdd(Gx + off + 1, w * gc.y);
  }
}

// ------------- combine + self-loop fold + re-zero accumulators -------------

__global__ void combine_kernel(const float* __restrict__ h, const float* __restrict__ xk,
                               const float* __restrict__ dsl, float* __restrict__ Ax,
                               float* __restrict__ Gx, float* __restrict__ g,
                               float* __restrict__ xk_out, int N) {
  const int idx = blockIdx.x * blockDim.x + threadIdx.x;  // one float2 each
  if (idx >= N * 32) return;
  const int node = idx >> 5;
  float ws = dsl[node];
  ws *= ws;                       // self-loop weight dinv_sl^2
  const size_t off = (size_t)idx * 2;
  const float2 xv = *(const float2*)(xk + off);
  const float2 hv = *(const float2*)(h + off);
  const float2 av = *(const float2*)(Ax + off);
  const float2 gxv = *(const float2*)(Gx + off);
  const float2 gv = *(const float2*)(g + off);
  float2 o;
  o.x = ALPHA * hv.x + REST * xv.x + BETA * (av.x + ws * xv.x)
        + (BETA * GAMMA) * (gxv.x + ws * gv.x);
  o.y = ALPHA * hv.y + REST * xv.y + BETA * (av.y + ws * xv.y)
        + (BETA * GAMMA) * (gxv.y + ws * gv.y);
  *(float2*)(xk_out + off) = o;
  const float2 z = {0.0f, 0.0f};   // re-zero accumulators for next iteration
  *(float2*)(Ax + off) = z;
  *(float2*)(Gx + off) = z;
  *(float2*)(g + off) = z;
}

// ------------- fused logits + log_softmax (one wave per row) ---------------

__global__ void __launch_bounds__(256) logits_kernel(
    const float* __restrict__ xk, const float* __restrict__ W3,
    float* __restrict__ out, int N) {
  __shared__ float rows[8][FDIM];
  const int wave = threadIdx.x >> 5;
  const int lane = threadIdx.x & 31;
  const int rid = blockIdx.x * 8 + wave;
  if (rid >= N) return;

  float2 v = *(const float2*)(xk + (size_t)rid * FDIM + lane * 2);
  v.x = v.x > 0.0f ? v.x : 0.0f;
  v.y = v.y > 0.0f ? v.y : 0.0f;
  rows[wave][lane * 2] = v.x;
  rows[wave][lane * 2 + 1] = v.y;
  asm volatile("s_wait_dscnt 0x0" ::: "memory");

  const int c0 = lane * 2, c1 = lane * 2 + 1;
  const int cc0 = c0 < CDIM ? c0 : CDIM - 1;   // clamp to stay in bounds
  const int cc1 = c1 < CDIM ? c1 : CDIM - 1;
  float a0 = 0.0f, a1 = 0.0f;
#pragma unroll 8
  for (int k = 0; k < FDIM; ++k) {
    const float xb = rows[wave][k];            // LDS broadcast read
    a0 += xb * W3[k * CDIM + cc0];
    a1 += xb * W3[k * CDIM + cc1];
  }
  float l0 = (c0 < CDIM) ? a0 : -INFINITY;
  float l1 = (c1 < CDIM) ? a1 : -INFINITY;

  float mx = fmaxf(l0, l1);
#pragma unroll
  for (int off = 16; off > 0; off >>= 1) mx = fmaxf(mx, __shfl_xor(mx, off, 32));
  float s = __expf(l0 - mx) + __expf(l1 - mx);  // exp(-inf)=0 for padding lanes
#pragma unroll
  for (int off = 16; off > 0; off >>= 1) s += __shfl_xor(s, off, 32);
  const float lse = mx + __logf(s);

  if (c0 < CDIM) out[(size_t)rid * CDIM + c0] = l0 - lse;
  if (c1 < CDIM) out[(size_t)rid * CDIM + c1] = l1 - lse;
}

// --------------------------------- launch ----------------------------------

extern "C" void kernel_launch(void* const* d_in, const int* in_sizes, int n_in,
                              void* d_out, int out_size, void* d_ws, size_t ws_size,
                              hipStream_t stream) {
  const float* x = (const float*)d_in[0];
  const int* ei = (const int*)d_in[1];
  const float* W1 = (const float*)d_in[2];
  const float* W2 = (const float*)d_in[3];
  const float* W3 = (const float*)d_in[4];

  const int N = in_sizes[0] / FDIM;
  const int E = in_sizes[1] / 2;
  const int* row = ei;
  const int* col = ei + E;

  // Workspace layout (floats)
  float* ws = (float*)d_ws;
  const size_t nf = (size_t)N * FDIM;
  float* h = ws;
  float* xkA = h + nf;
  float* xkB = xkA + nf;
  float* g = xkB + nf;         // g, Ax, Gx, deg contiguous for single zero pass
  float* Ax = g + nf;
  float* Gx = Ax + nf;
  float* deg = Gx + nf;
  float* dns = deg + N;
  float* dsl = dns + N;
  float* wns = dsl + N;
  float* wsl = wns + (size_t)E;

  // 1) normalization weights
  zero_kernel<<<2048, 256, 0, stream>>>(g, nf * 3 + (size_t)N);
  degree_kernel<<<(E + 255) / 256, 256, 0, stream>>>(col, deg, E);
  dinv_kernel<<<(N + 255) / 256, 256, 0, stream>>>(deg, dns, dsl, N);
  weights_kernel<<<(E + 255) / 256, 256, 0, stream>>>(row, col, dns, dsl, wns, wsl, E);

  // 2) MLP via WMMA
  mlp_kernel<<<(N + 63) / 64, 128, 0, stream>>>(x, W1, W2, h, N);

  // 3) K=4 diffusion steps (ping-pong xk buffers, xk0 = h)
  const float* xk = h;
  float* bufs[2] = {xkA, xkB};
  for (int it = 0; it < 4; ++it) {
    grad_kernel<<<8192, 256, 0, stream>>>(xk, row, col, wns, g, E);
    spmm_kernel<<<8192, 256, 0, stream>>>(xk, g, row, col, wsl, Ax, Gx, E);
    combine_kernel<<<(N * 32 + 255) / 256, 256, 0, stream>>>(h, xk, dsl, Ax, Gx, g,
                                                             bufs[it & 1], N);
    xk = bufs[it & 1];
  }

  // 4) logits + log_softmax
  logits_kernel<<<(N + 7) / 8, 256, 0, stream>>>(xk, W3, (float*)d_out, N);
}